// HybridMaxKSAGE_2259152798564
// MI455X (gfx1250) — compile-verified
//
#include <hip/hip_runtime.h>
#include <hip/hip_bf16.h>

typedef __attribute__((ext_vector_type(2))) float v2f;
typedef __attribute__((ext_vector_type(8))) float v8f;

#define N_NODES 100000
#define N_EDGES 1000000
#define IN_SIZE 128
#define HID 64
#define MAXK 16

// ---------------------------------------------------------------------------
// Degree kernels
// ---------------------------------------------------------------------------
__global__ void deg_count_kernel(const int* __restrict__ dst,
                                 float* __restrict__ deg, int nedges) {
    int e = blockIdx.x * blockDim.x + threadIdx.x;
    if (e < nedges) atomicAdd(&deg[dst[e]], 1.0f);
}

__global__ void deg_inv_kernel(float* __restrict__ deg, int n) {
    int i = blockIdx.x * blockDim.x + threadIdx.x;
    if (i < n) deg[i] = 1.0f / fmaxf(deg[i], 1.0f);
}

// ---------------------------------------------------------------------------
// WMMA fp32 GEMM: out[M,64] = A[M,K] @ W[64,K]^T (+ bias)
//                 (+ optional (A2[M,64] * rscale[row]) @ W2[64,64]^T)
// Block = 128 threads (4 waves). Each block: one 16-row M tile.
// Wave w handles output columns [16w, 16w+16).
// A fragment (16x4 f32): lanes 0-15 row=lane, K pair {k,k+1};
//                        lanes 16-31 row=lane-16, K pair {k+2,k+3}.
// B fragment (4x16 f32): symmetric, lane%16 = output column, same K split.
// W is row-major [64,K] == B^T, so B fragment loads are plain float2 reads.
// C/D: lanes 0-15 col=lane, vgpr r = row r; lanes 16-31 col=lane-16, row 8+r.
// ---------------------------------------------------------------------------
__global__ void gemm_wmma_kernel(const float* __restrict__ A, int lda, int K,
                                 const float* __restrict__ W,
                                 const float* __restrict__ bias,
                                 const float* __restrict__ A2,
                                 const float* __restrict__ W2,
                                 const float* __restrict__ rscale,
                                 float* __restrict__ out) {
    const int lane  = threadIdx.x & 31;
    const int waveN = threadIdx.x >> 5;          // 0..3 -> N tile
    const int row0  = blockIdx.x * 16;
    const int half  = lane >> 4;                 // K-pair select
    const int lm    = lane & 15;
    const int arow  = row0 + lm;                 // A row held by this lane
    const int wrow  = waveN * 16 + lm;           // weight row == output column

    v8f acc = {};

    const float* ap = A + (size_t)arow * lda + half * 2;
    const float* wp = W + (size_t)wrow * K + half * 2;
#pragma unroll 4
    for (int k = 0; k < K; k += 4) {
        v2f a = *(const v2f*)(ap + k);
        v2f b = *(const v2f*)(wp + k);
        acc = __builtin_amdgcn_wmma_f32_16x16x4_f32(
            false, a, false, b, (short)0, acc, false, false);
    }

    if (A2) {
        const float s = rscale[arow];
        const float* ap2 = A2 + (size_t)arow * HID + half * 2;
        const float* wp2 = W2 + (size_t)wrow * HID + half * 2;
#pragma unroll 4
        for (int k = 0; k < HID; k += 4) {
            v2f a = *(const v2f*)(ap2 + k);
            a = a * s;                            // per-row 1/deg scaling
            v2f b = *(const v2f*)(wp2 + k);
            acc = __builtin_amdgcn_wmma_f32_16x16x4_f32(
                false, a, false, b, (short)0, acc, false, false);
        }
    }

    const int col = waveN * 16 + lm;
    const float bv = bias ? bias[col] : 0.0f;
    float* op = out + (size_t)row0 * HID + col;
#pragma unroll
    for (int r = 0; r < 8; ++r) {
        int m = r + half * 8;
        op[(size_t)m * HID] = acc[r] + bv;
    }
}

// ---------------------------------------------------------------------------
// MaxK: one wave32 per row of 64; keep top-16, zero the rest.
// 16 rounds of wave-wide max + ballot; exact selection (ties by slot order).
// ---------------------------------------------------------------------------
__global__ void maxk_kernel(const float* __restrict__ h,
                            float* __restrict__ hs, int n) {
    const int wave = (blockIdx.x * blockDim.x + threadIdx.x) >> 5;
    const int lane = threadIdx.x & 31;
    if (wave >= n) return;

    const float* row = h + (size_t)wave * HID;
    const float a0 = row[lane];
    const float a1 = row[lane + 32];
    float r0 = a0, r1 = a1;
    bool k0 = false, k1 = false;
    const float NEG = -3.4e38f;

    for (int it = 0; it < MAXK; ++it) {
        float m = fmaxf(r0, r1);
        float g = m;
#pragma unroll
        for (int off = 16; off; off >>= 1)
            g = fmaxf(g, __shfl_xor(g, off));
        unsigned long long ball = __ballot(m == g);
        int first = __ffsll((long long)ball) - 1;
        if (lane == first) {
            if (r0 == g) { k0 = true; r0 = NEG; }
            else         { k1 = true; r1 = NEG; }
        }
    }

    float* o = hs + (size_t)wave * HID;
    o[lane]      = k0 ? a0 : 0.0f;
    o[lane + 32] = k1 ? a1 : 0.0f;
}

// ---------------------------------------------------------------------------
// Edge scatter: msum[dst] += hs[src], thread per (edge, col).
// hs is 16/64 sparse after MaxK -> skip zero atomics (~4x less atomic traffic).
// ---------------------------------------------------------------------------
__global__ void scatter_kernel(const float* __restrict__ hs,
                               const int* __restrict__ src,
                               const int* __restrict__ dst,
                               float* __restrict__ msum, int nedges) {
    const int t = blockIdx.x * blockDim.x + threadIdx.x;
    const int e = t >> 6;
    const int j = t & 63;
    if (e >= nedges) return;
    const float v = hs[(size_t)src[e] * HID + j];
    if (v != 0.0f) atomicAdd(&msum[(size_t)dst[e] * HID + j], v);
}

// ---------------------------------------------------------------------------
// Launch. d_in order: x, w_in, b_in, w_self, b_self, w_neigh, w_out, b_out,
//                     src, dst.   Workspace need: ~77.2 MB.
// ---------------------------------------------------------------------------
extern "C" void kernel_launch(void* const* d_in, const int* in_sizes, int n_in,
                              void* d_out, int out_size, void* d_ws, size_t ws_size,
                              hipStream_t stream) {
    const float* x       = (const float*)d_in[0];
    const float* w_in    = (const float*)d_in[1];
    const float* b_in    = (const float*)d_in[2];
    const float* w_self  = (const float*)d_in[3];   // [3,64,64]
    const float* b_self  = (const float*)d_in[4];   // [3,64]
    const float* w_neigh = (const float*)d_in[5];   // [3,64,64]
    const float* w_out   = (const float*)d_in[6];
    const float* b_out   = (const float*)d_in[7];
    const int*   src     = (const int*)d_in[8];
    const int*   dst     = (const int*)d_in[9];
    float* out = (float*)d_out;

    const int N = N_NODES, E = N_EDGES;

    // workspace layout
    float* deg  = (float*)d_ws;               // N        (becomes 1/deg)
    float* h    = deg + N;                    // N*64
    float* hs   = h + (size_t)N * HID;        // N*64
    float* msum = hs + (size_t)N * HID;       // N*64

    // degree -> 1/max(deg,1)
    hipMemsetAsync(deg, 0, (size_t)N * sizeof(float), stream);
    deg_count_kernel<<<(E + 255) / 256, 256, 0, stream>>>(dst, deg, E);
    deg_inv_kernel<<<(N + 255) / 256, 256, 0, stream>>>(deg, N);

    // h = x @ w_in^T + b_in   (M=100000, N=64, K=128)
    gemm_wmma_kernel<<<N / 16, 128, 0, stream>>>(
        x, IN_SIZE, IN_SIZE, w_in, b_in, nullptr, nullptr, nullptr, h);

    for (int i = 0; i < 3; ++i) {
        // hs = maxk(h, 16)
        maxk_kernel<<<(N + 7) / 8, 256, 0, stream>>>(h, hs, N);
        // msum = segment_sum(hs[src], dst)
        hipMemsetAsync(msum, 0, (size_t)N * HID * sizeof(float), stream);
        scatter_kernel<<<(E * 64) / 256, 256, 0, stream>>>(hs, src, dst, msum, E);
        // h = hs @ w_self[i]^T + b_self[i] + (msum/deg) @ w_neigh[i]^T
        gemm_wmma_kernel<<<N / 16, 128, 0, stream>>>(
            hs, HID, HID,
            w_self + (size_t)i * HID * HID, b_self + (size_t)i * HID,
            msum, w_neigh + (size_t)i * HID * HID, deg, h);
    }

    // out = h @ w_out^T + b_out
    gemm_wmma_kernel<<<N / 16, 128, 0, stream>>>(
        h, HID, HID, w_out, b_out, nullptr, nullptr, nullptr, out);
}